// NodeNet_24936580121171
// MI455X (gfx1250) — compile-verified
//
#include <hip/hip_runtime.h>
#include <hip/hip_bf16.h>
#include <math.h>

typedef __attribute__((ext_vector_type(2))) float v2f;
typedef __attribute__((ext_vector_type(4))) float v4f;
typedef __attribute__((ext_vector_type(8))) float v8f;

// Branchless fast activations: hardware v_exp_f32 + raw v_rcp_f32
// (no IEEE divide refinement sequence).  Limits are exact:
// expf->inf => tanh=1, expf->0 => tanh=-1.
__device__ __forceinline__ float rcp_fast(float x) {
    return __builtin_amdgcn_rcpf(x);
}
__device__ __forceinline__ float tanh_fast(float x) {
    return 1.0f - 2.0f * rcp_fast(__expf(2.0f * x) + 1.0f);
}
__device__ __forceinline__ float sigmoid_fast(float x) {
    return rcp_fast(1.0f + __expf(-x));
}

// ---------------------------------------------------------------------------
// Kernel 0: zero the workspace accumulators (idxf_i, idxf_o, mi, mo).
// Required every call: harness poisons ws once and never re-poisons.
// ---------------------------------------------------------------------------
__global__ void zero_ws_kernel(float* ws, int count) {
    int i = blockIdx.x * blockDim.x + threadIdx.x;
    int stride = gridDim.x * blockDim.x;
    for (; i < count; i += stride) ws[i] = 0.0f;
}

// ---------------------------------------------------------------------------
// Kernel 0b: zero-pad the MLP weights to Hp columns so the WMMA kernel's
// B-fragment / bias / W2 loads are unconditional (no EXEC-mask branching).
// ---------------------------------------------------------------------------
__global__ void pad_weights_kernel(const float* __restrict__ W1,
                                   const float* __restrict__ b1,
                                   const float* __restrict__ W2,
                                   float* w1p, float* b1p, float* w2p,
                                   int H, int Hp, int K) {
    const int i = blockIdx.x * blockDim.x + threadIdx.x;
    const int tot = K * Hp;
    if (i < tot) {
        const int r = i / Hp, c = i % Hp;
        w1p[i] = (c < H) ? W1[r * H + c] : 0.0f;
    }
    if (i < Hp) {
        b1p[i] = (i < H) ? b1[i] : 0.0f;
        w2p[i] = (i < H) ? W2[i] : 0.0f;
    }
}

// ---------------------------------------------------------------------------
// Kernel 1: streaming sparse-index extraction from the dense one-hot
// incidence matrices (the ONLY pass over the 1.6 GB of data => HBM-bound,
// ~69 us floor at 23.3 TB/s).  Each thread owns 4 consecutive columns and
// streams b128 nontemporal loads (512 B per wave per issue) over a row
// chunk of both Ri and Ro, accumulating sum(n * R[n,k]) per column.
// One-hot => at most one chunk yields a nonzero partial, posted with a
// single exact f32 atomicAdd.
// ---------------------------------------------------------------------------
__global__ void extract_kernel(const float* __restrict__ Ri,
                               const float* __restrict__ Ro,
                               float* idxf_i, float* idxf_o,
                               int E, int rows_per_chunk) {
    const int  k4  = blockIdx.x * blockDim.x + threadIdx.x;  // 4-column group
    const long col = (long)k4 * 4;
    const long n0  = (long)blockIdx.y * rows_per_chunk;
    v4f si = {0.0f, 0.0f, 0.0f, 0.0f};
    v4f so = {0.0f, 0.0f, 0.0f, 0.0f};
    #pragma unroll 4
    for (int r = 0; r < rows_per_chunk; ++r) {
        const long n = n0 + r;
        const v4f ri = __builtin_nontemporal_load(
            (const v4f*)(Ri + n * (long)E + col));
        const v4f ro = __builtin_nontemporal_load(
            (const v4f*)(Ro + n * (long)E + col));
        const float nf = (float)n;
        si += ri * nf;
        so += ro * nf;
    }
    #pragma unroll
    for (int j = 0; j < 4; ++j) {
        if (si[j] != 0.0f) atomicAdd(idxf_i + col + j, si[j]);
        if (so[j] != 0.0f) atomicAdd(idxf_o + col + j, so[j]);
    }
}

// ---------------------------------------------------------------------------
// Kernel 2: per-edge gather + weighted scatter-add.
//   mi[idx_i[k]] += e[k] * X[idx_o[k]]
//   mo[idx_o[k]] += e[k] * X[idx_i[k]]
// ---------------------------------------------------------------------------
__global__ void scatter_kernel(const float* __restrict__ idxf_i,
                               const float* __restrict__ idxf_o,
                               const float* __restrict__ e,
                               const float* __restrict__ X,
                               float* mi, float* mo, int E) {
    const int k = blockIdx.x * blockDim.x + threadIdx.x;
    if (k >= E) return;
    const int ii = (int)(idxf_i[k] + 0.5f);
    const int io = (int)(idxf_o[k] + 0.5f);
    const float w = e[k];
    #pragma unroll
    for (int d = 0; d < 4; ++d) {
        atomicAdd(&mi[ii * 4 + d], w * X[io * 4 + d]);
        atomicAdd(&mo[io * 4 + d], w * X[ii * 4 + d]);
    }
}

// ---------------------------------------------------------------------------
// Kernel 3: per-node MLP via V_WMMA_F32_16X16X4_F32 (wave32).
// M[n, 0:12] = [mi | mo | X], layer1: tanh(M @ W1 + b1)  (K=12 = 3 x K=4),
// layer2: sigmoid(h @ W2 + b2) reduced across lanes with shfl_xor.
// One wave computes 16 nodes; 7 hidden tiles of 16 cover Hp=112 (zero-padded).
//
// ISA layouts (CDNA5 ISA):
//   A (16x4 f32):  lane l, elem j  -> A[l%16, 2*(l/16)+j]
//   B (4x16 f32):  lane l, elem j  -> B[2*(l/16)+j, l%16]
//   C/D (16x16):   lane l, vgpr v  -> D[v + 8*(l/16), l%16]
// ---------------------------------------------------------------------------
__global__ void mlp_wmma_kernel(const float* __restrict__ mi,
                                const float* __restrict__ mo,
                                const float* __restrict__ X,
                                const float* __restrict__ w1p,  // [12, Hp]
                                const float* __restrict__ b1p,  // [Hp]
                                const float* __restrict__ w2p,  // [Hp]
                                const float* __restrict__ b2,
                                float* __restrict__ out, int Hp) {
    const int lane = threadIdx.x & 31;
    const int wave = (blockIdx.x * blockDim.x + threadIdx.x) >> 5;
    const int r0   = wave * 16;        // first node of this wave's tile
    const int g    = lane >> 4;        // lane half
    const int m    = lane & 15;        // position within half
    const int row  = r0 + m;           // node this lane's A-fragment covers

    // A fragments for the three K=4 chunks (columns 0..11 of M).
    // Branchless: select base pointer, offset is (col & 3).
    v2f a[3];
    #pragma unroll
    for (int c = 0; c < 3; ++c) {
        #pragma unroll
        for (int j = 0; j < 2; ++j) {
            const int col = 4 * c + 2 * g + j;
            const float* src = (col < 4) ? mi : ((col < 8) ? mo : X);
            a[c][j] = src[row * 4 + (col & 3)];
        }
    }

    float acc2[8];
    #pragma unroll
    for (int v = 0; v < 8; ++v) acc2[v] = 0.0f;

    const int ntiles = Hp / 16;         // 7 for Hp=112
    for (int t = 0; t < ntiles; ++t) {
        const int hcol = 16 * t + m;    // hidden unit for this lane's column
        v8f cfrag = {};
        #pragma unroll
        for (int c = 0; c < 3; ++c) {
            v2f b;
            #pragma unroll
            for (int j = 0; j < 2; ++j) {
                const int k = 4 * c + 2 * g + j;      // input-feature row
                b[j] = w1p[k * Hp + hcol];            // unconditional load
            }
            cfrag = __builtin_amdgcn_wmma_f32_16x16x4_f32(
                false, a[c], false, b, (short)0, cfrag, false, false);
        }
        const float bias = b1p[hcol];
        const float w2   = w2p[hcol];
        #pragma unroll
        for (int v = 0; v < 8; ++v) {
            // Branchless tanh (v_exp + v_rcp); padded columns give 0 * 0.
            acc2[v] += tanh_fast(cfrag[v] + bias) * w2;
        }
    }

    // Reduce layer-2 partials across the 16 lanes of each half
    // (xor masks < 16 stay inside the half). Lane m==0 of each half holds
    // the full dot product for node r0 + v + 8*g.
    const float bb = b2[0];
    #pragma unroll
    for (int v = 0; v < 8; ++v) {
        float s = acc2[v];
        s += __shfl_xor(s, 1, 32);
        s += __shfl_xor(s, 2, 32);
        s += __shfl_xor(s, 4, 32);
        s += __shfl_xor(s, 8, 32);
        if (m == 0)
            out[r0 + v + 8 * g] = sigmoid_fast(s + bb);
    }
}

// ---------------------------------------------------------------------------
extern "C" void kernel_launch(void* const* d_in, const int* in_sizes, int n_in,
                              void* d_out, int out_size, void* d_ws, size_t ws_size,
                              hipStream_t stream) {
    const float* X  = (const float*)d_in[0];   // [N, 4]
    const float* e  = (const float*)d_in[1];   // [E, 1]
    const float* Ri = (const float*)d_in[2];   // [N, E]
    const float* Ro = (const float*)d_in[3];   // [N, E]
    const float* W1 = (const float*)d_in[4];   // [12, H]
    const float* b1 = (const float*)d_in[5];   // [H]
    const float* W2 = (const float*)d_in[6];   // [H, 1]
    const float* b2 = (const float*)d_in[7];   // [1]

    const int N  = in_sizes[0] / 4;            // 8192
    const int E  = in_sizes[1];                // 24576
    const int H  = in_sizes[5];                // 100
    const int K  = 12;                         // 3*D MLP input width
    const int Hp = ((H + 15) / 16) * 16;       // 112

    // Workspace layout (floats):
    //   idxf_i[E] | idxf_o[E] | mi[4N] | mo[4N] | w1p[K*Hp] | b1p[Hp] | w2p[Hp]
    float* ws     = (float*)d_ws;
    float* idxf_i = ws;
    float* idxf_o = ws + E;
    float* mi     = ws + 2 * E;
    float* mo     = ws + 2 * E + 4 * N;
    float* w1p    = ws + 2 * E + 8 * N;
    float* b1p    = w1p + K * Hp;
    float* w2p    = b1p + Hp;
    const int zcount = 2 * E + 8 * N;          // accumulators only

    zero_ws_kernel<<<256, 256, 0, stream>>>(ws, zcount);
    pad_weights_kernel<<<(K * Hp + 255) / 256, 256, 0, stream>>>(
        W1, b1, W2, w1p, b1p, w2p, H, Hp, K);

    const int chunks = 32;                     // row chunks for parallelism
    const int rpc    = N / chunks;             // 256 rows per chunk
    dim3 egrid((E / 4) / 256, chunks);         // 4 columns per thread (b128)
    extract_kernel<<<egrid, 256, 0, stream>>>(Ri, Ro, idxf_i, idxf_o, E, rpc);

    scatter_kernel<<<(E + 255) / 256, 256, 0, stream>>>(idxf_i, idxf_o, e, X,
                                                        mi, mo, E);

    const int waves = N / 16;                  // 512 node tiles
    mlp_wmma_kernel<<<waves / 4, 128, 0, stream>>>(mi, mo, X, w1p, b1p, w2p,
                                                   b2, (float*)d_out, Hp);
}